// NeuralODE_17454747090999
// MI455X (gfx1250) — compile-verified
//
#include <hip/hip_runtime.h>

#define NWG    64
#define TPB    256
#define NBATCH 8192
#define NT     1001
#define RTOL   1.0e-3f
#define ATOL   1.0e-4f

typedef float v2f __attribute__((ext_vector_type(2)));
typedef float v8f __attribute__((ext_vector_type(8)));

// Swap lane halves 0-15 <-> 16-31 (ds_swizzle group-of-32: xor=0x10, and=0x1f).
// Inline asm (volatile) so the compiler cannot sink it into a predicated
// region: ds_swizzle reads disabled lanes as zero, so it MUST run with full
// EXEC. Waiting dscnt<=0 is conservative w.r.t. compiler DS bookkeeping.
__device__ __forceinline__ float swap16(float x) {
  float r;
  asm volatile("ds_swizzle_b32 %0, %1 offset:0x401f\n\t"
               "s_wait_dscnt 0x0"
               : "=v"(r)
               : "v"(x));
  return r;
}

// Batched version: issue 8 swizzles back-to-back, wait once. The DS pipe keeps
// all 8 in flight, so one latency is paid instead of eight on the critical
// sequential dopri5 chain. Early-clobber outputs: results must not alias the
// source operands of the later swizzles in the same block.
__device__ __forceinline__ void swap16x8(const float* z, float* w) {
  asm volatile(
      "ds_swizzle_b32 %0, %8 offset:0x401f\n\t"
      "ds_swizzle_b32 %1, %9 offset:0x401f\n\t"
      "ds_swizzle_b32 %2, %10 offset:0x401f\n\t"
      "ds_swizzle_b32 %3, %11 offset:0x401f\n\t"
      "ds_swizzle_b32 %4, %12 offset:0x401f\n\t"
      "ds_swizzle_b32 %5, %13 offset:0x401f\n\t"
      "ds_swizzle_b32 %6, %14 offset:0x401f\n\t"
      "ds_swizzle_b32 %7, %15 offset:0x401f\n\t"
      "s_wait_dscnt 0x0"
      : "=&v"(w[0]), "=&v"(w[1]), "=&v"(w[2]), "=&v"(w[3]),
        "=&v"(w[4]), "=&v"(w[5]), "=&v"(w[6]), "=&v"(w[7])
      : "v"(z[0]), "v"(z[1]), "v"(z[2]), "v"(z[3]),
        "v"(z[4]), "v"(z[5]), "v"(z[6]), "v"(z[7]));
}

__device__ __forceinline__ v8f wmma4(v2f a, v2f b, v8f c) {
  // D = A(16x4 f32) x B(4x16 f32) + C(16x16 f32)
  return __builtin_amdgcn_wmma_f32_16x16x4_f32(false, a, false, b, (short)0, c,
                                               false, false);
}
__device__ __forceinline__ float ftanh(float x) {
  float e = __expf(2.0f * x);                       // v_exp_f32
  return 1.0f - 2.0f * __builtin_amdgcn_rcpf(e + 1.0f);
}
__device__ __forceinline__ float uniformf(float x) {
  return __int_as_float(__builtin_amdgcn_readfirstlane(__float_as_int(x)));
}

struct Consts {
  v2f   A1[2];    // W1^T as two 16x4 A-tiles (hidden rows 0-15 / 16-31)
  v2f   A2[8];    // W2^T as eight 16x4 A-tiles (K chunks of the 32 hidden)
  float b1r[16];  // per-lane hidden bias in C-layout
  float bo0, bo1; // output bias in state layout
  bool  lo;       // lane < 16
};

// f(y) = tanh(y@W1+b1)@W2 + b2, computed transposed so the wave's state layout
// (B-matrix 4x16) feeds WMMA directly with no data movement on the input side.
__device__ __forceinline__ v2f evalf(v2f y, const Consts& C) {
  v8f c0 = {}, c1 = {};
  c0 = wmma4(C.A1[0], y, c0);   // hidden rows 0-15  x 16 batch cols
  c1 = wmma4(C.A1[1], y, c1);   // hidden rows 16-31 x 16 batch cols
  float z[16];
#pragma unroll
  for (int r = 0; r < 8; ++r) z[r]     = ftanh(c0[r] + C.b1r[r]);
#pragma unroll
  for (int r = 0; r < 8; ++r) z[8 + r] = ftanh(c1[r] + C.b1r[8 + r]);

  // Every z register is needed once in swapped form: batch-swizzle per tile.
  float w[16];
  swap16x8(z, w);
  swap16x8(z + 8, w + 8);

  v8f acc = {};
#pragma unroll
  for (int c = 0; c < 8; ++c) {         // K = 4c..4c+3 of z^T (32x16)
    const int cc = c & 3;
    const int zb = (c >> 2) * 8 + (cc & 1) * 4;  // base VGPR of the 4 K rows
    const int up = cc >> 1;                      // K rows live in upper halves?
    float bv0, bv1;
    if (up == 0) {                     // sources in lanes 0-15 of z regs
      bv0 = C.lo ? z[zb]     : w[zb + 2];  // B.v0: lo=K0, hi=K2
      bv1 = C.lo ? z[zb + 1] : w[zb + 3];  // B.v1: lo=K1, hi=K3
    } else {                           // sources in lanes 16-31 of z regs
      bv0 = C.lo ? w[zb]     : z[zb + 2];
      bv1 = C.lo ? w[zb + 1] : z[zb + 3];
    }
    v2f B; B.x = bv0; B.y = bv1;
    acc = wmma4(C.A2[c], B, acc);
  }
  // acc rows 0..2 = out dims 0..2 per batch lane (upper halves are zero pad).
  float a2s = swap16(acc[2]);          // dim2 moved to upper half lanes
  v2f k;
  k.x = (C.lo ? acc[0] : a2s) + C.bo0; // lo: d0, hi: d2
  k.y = acc[1] + C.bo1;                // lo: d1, hi: 0
  return k;
}

__global__ void node_init_kernel(unsigned* bar) {
  if (threadIdx.x == 0) *bar = 0u;
}

__global__ __launch_bounds__(TPB) void node_dopri5_kernel(
    const float* __restrict__ u0, const float* __restrict__ W1,
    const float* __restrict__ b1, const float* __restrict__ W2,
    const float* __restrict__ b2, const float* __restrict__ t,
    float* __restrict__ out, unsigned* __restrict__ bar,
    float* __restrict__ gpart) {
  __shared__ float lds_part[TPB / 32];
  __shared__ float lds_total;

  const int  lane   = threadIdx.x & 31;
  const int  waveId = threadIdx.x >> 5;
  const int  gwave  = blockIdx.x * (TPB / 32) + waveId;  // 0..511
  const int  rowB   = gwave * 16;
  const bool lo     = lane < 16;
  const int  m      = lo ? lane : lane - 16;

  // ---- preload constant matrices into WMMA A-layouts (registers) ----
  Consts C;
  C.lo = lo;
#pragma unroll
  for (int tt = 0; tt < 2; ++tt) {
    int hid = tt * 16 + m;
    float w0 = W1[0 * 32 + hid], w1 = W1[1 * 32 + hid], w2 = W1[2 * 32 + hid];
    C.A1[tt].x = lo ? w0 : w2;
    C.A1[tt].y = lo ? w1 : 0.0f;
  }
#pragma unroll
  for (int c = 0; c < 8; ++c) {
    int k0 = lo ? 4 * c + 0 : 4 * c + 2;
    int k1 = lo ? 4 * c + 1 : 4 * c + 3;
    float wa = W2[min(k0 * 3 + m, 95)];
    float wb = W2[min(k1 * 3 + m, 95)];
    C.A2[c].x = (m < 3) ? wa : 0.0f;
    C.A2[c].y = (m < 3) ? wb : 0.0f;
  }
#pragma unroll
  for (int tt = 0; tt < 2; ++tt)
#pragma unroll
    for (int r = 0; r < 8; ++r) {
      float blo = b1[tt * 16 + r], bhi = b1[tt * 16 + r + 8];
      C.b1r[tt * 8 + r] = lo ? blo : bhi;
    }
  C.bo0 = lo ? b2[0] : b2[2];
  C.bo1 = lo ? b2[1] : 0.0f;

  // ---- load initial state into B-layout: lo lane = (d0,d1), hi = (d2,0) ----
  v2f y;
  {
    int row = rowB + m;
    float a = u0[row * 3 + 0], bb = u0[row * 3 + 1], cz = u0[row * 3 + 2];
    y.x = lo ? a : cz;
    y.y = lo ? bb : 0.0f;
  }

  // dopri5 tableau (fp32)
  const float A21 = (float)(1.0 / 5.0);
  const float A31 = (float)(3.0 / 40.0),  A32 = (float)(9.0 / 40.0);
  const float A41 = (float)(44.0 / 45.0), A42 = (float)(-56.0 / 15.0), A43 = (float)(32.0 / 9.0);
  const float A51 = (float)(19372.0 / 6561.0), A52 = (float)(-25360.0 / 2187.0),
              A53 = (float)(64448.0 / 6561.0), A54 = (float)(-212.0 / 729.0);
  const float A61 = (float)(9017.0 / 3168.0),  A62 = (float)(-355.0 / 33.0),
              A63 = (float)(46732.0 / 5247.0), A64 = (float)(49.0 / 176.0),
              A65 = (float)(-5103.0 / 18656.0);
  const float B1c = (float)(35.0 / 384.0), B3c = (float)(500.0 / 1113.0),
              B4c = (float)(125.0 / 192.0), B5c = (float)(-2187.0 / 6784.0),
              B6c = (float)(11.0 / 84.0);
  const float E1 = (float)(35.0 / 384.0 - 5179.0 / 57600.0);
  const float E3 = (float)(500.0 / 1113.0 - 7571.0 / 16695.0);
  const float E4 = (float)(125.0 / 192.0 - 393.0 / 640.0);
  const float E5 = (float)(-2187.0 / 6784.0 + 92097.0 / 339200.0);
  const float E6 = (float)(11.0 / 84.0 - 187.0 / 2100.0);
  const float E7 = (float)(-1.0 / 40.0);

  float tc = t[0];
  float h  = t[1] - t[0];
  unsigned gen = 0;

  // out[0] = u0
  {
    size_t base = (size_t)(rowB + m) * 3;
    if (lo) { out[base + 0] = y.x; out[base + 1] = y.y; }
    else    { out[base + 2] = y.x; }
  }

  for (int i = 1; i < NT; ++i) {
    float tn = uniformf(t[i]);
    for (int att = 0; att < 6; ++att) {
      float rem = tn - tc;
      if (!(rem > 0.0f)) continue;     // uniform across whole grid
      float heff = fminf(h, rem);

      v2f k1 = evalf(y, C);
      v2f s  = A21 * k1;
      v2f k2 = evalf(y + heff * s, C);
      s = A31 * k1; s += A32 * k2;
      v2f k3 = evalf(y + heff * s, C);
      s = A41 * k1; s += A42 * k2; s += A43 * k3;
      v2f k4 = evalf(y + heff * s, C);
      s = A51 * k1; s += A52 * k2; s += A53 * k3; s += A54 * k4;
      v2f k5 = evalf(y + heff * s, C);
      s = A61 * k1; s += A62 * k2; s += A63 * k3; s += A64 * k4; s += A65 * k5;
      v2f k6 = evalf(y + heff * s, C);
      s = B1c * k1; s += B3c * k3; s += B4c * k4; s += B5c * k5; s += B6c * k6;
      v2f y5 = y + heff * s;           // FSAL: stage-7 input == y5
      v2f k7 = evalf(y5, C);
      s = E1 * k1; s += E3 * k3; s += E4 * k4; s += E5 * k5; s += E6 * k6; s += E7 * k7;
      v2f err = heff * s;

      v2f sc;
      sc.x = ATOL + RTOL * fmaxf(fabsf(y.x), fabsf(y5.x));
      sc.y = ATOL + RTOL * fmaxf(fabsf(y.y), fabsf(y5.y));
      float r0 = err.x / sc.x, r1 = err.y / sc.y;
      float ss = r0 * r0 + r1 * r1;    // pad slots are exactly zero

      // wave butterfly reduce (32 lanes)
#pragma unroll
      for (int off = 16; off; off >>= 1) ss += __shfl_xor(ss, off, 32);
      if (lane == 0) lds_part[waveId] = ss;
      __syncthreads();

      const unsigned slot = gen & 1u;
      if (threadIdx.x == 0) {
        float w = 0.0f;
#pragma unroll
        for (int q = 0; q < TPB / 32; ++q) w += lds_part[q];
        __hip_atomic_store(&gpart[slot * NWG + blockIdx.x], w,
                           __ATOMIC_RELAXED, __HIP_MEMORY_SCOPE_AGENT);
      }
      gen++;
      __syncthreads();
      if (threadIdx.x == 0) {
        __threadfence();
        atomicAdd(bar, 1u);
        const unsigned target = gen * NWG;
        while (__hip_atomic_load(bar, __ATOMIC_ACQUIRE,
                                 __HIP_MEMORY_SCOPE_AGENT) < target)
          __builtin_amdgcn_s_sleep(2);
        float tot = 0.0f;
        for (int q = 0; q < NWG; ++q)  // fixed order -> deterministic
          tot += __hip_atomic_load(&gpart[slot * NWG + q], __ATOMIC_RELAXED,
                                   __HIP_MEMORY_SCOPE_AGENT);
        lds_total = tot;
      }
      __syncthreads();

      float total = uniformf(lds_total);
      float ratio = sqrtf(total * (1.0f / (float)(NBATCH * 3)));
      if (ratio <= 1.0f) { y = y5; tc = tc + heff; }
      float rmax = fmaxf(ratio, 1.0e-10f);
      float fac  = 0.9f * exp2f(-0.2f * log2f(rmax));
      fac = fminf(10.0f, fmaxf(0.2f, fac));
      h = h * fac;
    }
    // emit y at grid point i
    size_t base = ((size_t)i * NBATCH + (size_t)(rowB + m)) * 3;
    if (lo) { out[base + 0] = y.x; out[base + 1] = y.y; }
    else    { out[base + 2] = y.x; }
  }
}

extern "C" void kernel_launch(void* const* d_in, const int* in_sizes, int n_in,
                              void* d_out, int out_size, void* d_ws,
                              size_t ws_size, hipStream_t stream) {
  (void)in_sizes; (void)n_in; (void)out_size; (void)ws_size;
  const float* u0 = (const float*)d_in[0];
  const float* W1 = (const float*)d_in[1];
  const float* b1 = (const float*)d_in[2];
  const float* W2 = (const float*)d_in[3];
  const float* b2 = (const float*)d_in[4];
  const float* t  = (const float*)d_in[5];
  float* out = (float*)d_out;

  unsigned* bar  = (unsigned*)d_ws;                   // barrier counter
  float* gpart   = (float*)((char*)d_ws + 128);       // 2 x NWG partials

  hipLaunchKernelGGL(node_init_kernel, dim3(1), dim3(32), 0, stream, bar);
  hipLaunchKernelGGL(node_dopri5_kernel, dim3(NWG), dim3(TPB), 0, stream,
                     u0, W1, b1, W2, b2, t, out, bar, gpart);
}